// Left_Pool_66623532695930
// MI455X (gfx1250) — compile-verified
//
#include <hip/hip_runtime.h>

// ---------------------------------------------------------------------------
// Left pool (CornerNet) = reverse cummax along W.
// (32, 256, 4096) f32 -> 8192 rows x 4096. Memory-bound: 256 MiB @ 23.3 TB/s
// => ~11 us floor. Strategy: CDNA5 async DMA global->LDS, block-level suffix
// max scan in LDS (wave32-safe via __syncthreads), async DMA LDS->global.
// ---------------------------------------------------------------------------

#define __as1 __attribute__((address_space(1)))
#define __as3 __attribute__((address_space(3)))

typedef int v4i __attribute__((vector_size(16)));

__device__ __forceinline__ void wait_async0() {
#if __has_builtin(__builtin_amdgcn_s_wait_asynccnt)
  __builtin_amdgcn_s_wait_asynccnt(0);
#else
  asm volatile("s_wait_asynccnt 0" ::: "memory");
#endif
}

// Async copy 16 bytes global -> LDS (per-lane addresses). Tracked by ASYNCcnt.
__device__ __forceinline__ void stage_in_b128(const float* gsrc, float* ldst) {
#if __has_builtin(__builtin_amdgcn_global_load_async_to_lds_b128)
  __builtin_amdgcn_global_load_async_to_lds_b128((__as1 v4i*)gsrc,
                                                 (__as3 v4i*)ldst,
                                                 /*offset=*/0, /*cpol=*/0);
#else
  *(float4*)ldst = *(const float4*)gsrc;
#endif
}

// Async copy 16 bytes LDS -> global (per-lane addresses). Tracked by ASYNCcnt.
__device__ __forceinline__ void stage_out_b128(float* gdst, const float* lsrc) {
#if __has_builtin(__builtin_amdgcn_global_store_async_from_lds_b128)
  __builtin_amdgcn_global_store_async_from_lds_b128((__as1 v4i*)gdst,
                                                    (__as3 v4i*)lsrc,
                                                    /*offset=*/0, /*cpol=*/0);
#else
  *(float4*)gdst = *(const float4*)lsrc;
#endif
}

__global__ __launch_bounds__(256) void left_pool_kernel(
    const float* __restrict__ x, float* __restrict__ out) {
  constexpr int W = 4096;       // row length
  constexpr int T = 256;        // threads per block (8 wave32s)
  constexpr int CHUNK = W / T;  // 16 contiguous elements per thread

  __shared__ __align__(16) float row[W];   // 16 KB staging buffer
  __shared__ float smax[T];                // per-thread maxima for the scan

  const int tid = threadIdx.x;
  const long long base = (long long)blockIdx.x * W;
  const float* g = x + base;
  float* o = out + base;

  // ---- Stage row into LDS: coalesced 16B/lane async copies -----------------
#pragma unroll
  for (int k = 0; k < 4; ++k) {
    const int off = k * (T * 4) + tid * 4;  // lane-contiguous float4s
    stage_in_b128(g + off, &row[off]);
  }
  wait_async0();     // this wave's async loads have landed in LDS
  __syncthreads();   // all waves executed their waits -> whole row visible

  // ---- Each thread grabs its 16 contiguous elements + local max ------------
  const int cbase = tid * CHUNK;
  float r[CHUNK];
#pragma unroll
  for (int k = 0; k < CHUNK; ++k) r[k] = row[cbase + k];

  float lmax = r[0];
#pragma unroll
  for (int k = 1; k < CHUNK; ++k) lmax = fmaxf(lmax, r[k]);

  // ---- Block-wide suffix-max scan over the 256 thread maxima ---------------
  smax[tid] = lmax;
  __syncthreads();
  float v = lmax;
#pragma unroll
  for (int off = 1; off < T; off <<= 1) {
    const float othr = (tid + off < T) ? smax[tid + off] : -__builtin_inff();
    __syncthreads();
    v = fmaxf(v, othr);
    smax[tid] = v;
    __syncthreads();
  }
  // exclusive suffix (max of everything strictly to the right of this chunk)
  const float seed = (tid + 1 < T) ? smax[tid + 1] : -__builtin_inff();

  // ---- Sequential reverse cummax inside the chunk --------------------------
  float m = seed;
#pragma unroll
  for (int k = CHUNK - 1; k >= 0; --k) {
    m = fmaxf(r[k], m);
    r[k] = m;
  }

  // ---- Write results back to LDS, then coalesced async stores to global ----
#pragma unroll
  for (int k = 0; k < CHUNK; ++k) row[cbase + k] = r[k];
  __syncthreads();

#pragma unroll
  for (int k = 0; k < 4; ++k) {
    const int off = k * (T * 4) + tid * 4;
    stage_out_b128(o + off, &row[off]);
  }
  wait_async0();  // drain async stores before wave exit
}

extern "C" void kernel_launch(void* const* d_in, const int* in_sizes, int n_in,
                              void* d_out, int out_size, void* d_ws,
                              size_t ws_size, hipStream_t stream) {
  const float* x = (const float*)d_in[0];
  float* out = (float*)d_out;
  constexpr int W = 4096;
  const int rows = in_sizes[0] / W;  // 32 * 256 = 8192 rows
  left_pool_kernel<<<rows, 256, 0, stream>>>(x, out);
}